// MRConv2d_47863115547113
// MI455X (gfx1250) — compile-verified
//
#include <hip/hip_runtime.h>
#include <stdint.h>

// MRConv2d fused kernels for gfx1250 (MI455X).
//   K0: deinterleave W (OUT,2C) -> Wx (OUT,C), Wj (OUT,C)     [64 KB, one-time]
//   K1: transpose x (B,C,N) -> xT (B,N,C)                     [gathers become 256B rows]
//   K2: per (batch, 32-node tile):
//       - async-copy the contiguous x tile (32 x 256B) into LDS (GLOBAL_LOAD_ASYNC_TO_LDS_B128)
//       - overlapped: gather-max over K=16 neighbors -> xj tile in LDS
//       - s_wait_asynccnt 0 + barrier
//       - out = relu(Wx @ hx + Wj @ hj + bias) via V_WMMA_F32_16X16X4_F32 (exact fp32)

typedef __attribute__((ext_vector_type(2))) float v2f;
typedef __attribute__((ext_vector_type(8))) float v8f;
typedef int v4i __attribute__((vector_size(16)));

typedef __attribute__((address_space(1))) v4i* g_v4i_p;   // global (AS1) 16B vector ptr
typedef __attribute__((address_space(3))) v4i* l_v4i_p;   // LDS (AS3) 16B vector ptr

#define B_   4
#define C_   64
#define N_   8192
#define K_   16
#define OUT_ 128
#define TN   32          // nodes per block tile
#define HS2  68          // node-major LDS row stride in floats (64 + 4 pad; 272B rows, 16B aligned)

// ---------------- K0: deinterleave W columns ----------------
__global__ __launch_bounds__(256) void wprep_kernel(const float* __restrict__ W,
                                                    float* __restrict__ Wx,
                                                    float* __restrict__ Wj) {
  const int t = blockIdx.x * 256 + threadIdx.x;   // 0 .. OUT_*C_-1
  const int o = t >> 6, c = t & 63;
  Wx[t] = W[o * (2 * C_) + 2 * c + 0];
  Wj[t] = W[o * (2 * C_) + 2 * c + 1];
}

// ---------------- K1: tiled transpose (B,C,N) -> (B,N,C) ----------------
__global__ __launch_bounds__(256) void xpose_kernel(const float* __restrict__ x,
                                                    float* __restrict__ xT) {
  __shared__ float tile[32][33];
  const int b  = blockIdx.z;
  const int n0 = blockIdx.x * 32;
  const int c0 = blockIdx.y * 32;
  const int tx = threadIdx.x, ty = threadIdx.y;
#pragma unroll
  for (int i = 0; i < 32; i += 8)
    tile[ty + i][tx] = x[(b * C_ + (c0 + ty + i)) * N_ + n0 + tx];
  __syncthreads();
#pragma unroll
  for (int i = 0; i < 32; i += 8)
    xT[(b * N_ + (n0 + ty + i)) * C_ + c0 + tx] = tile[tx][ty + i];
}

// ---------------- K2: fused gather-max + WMMA GEMM + bias + relu ----------------
__global__ __launch_bounds__(256) void mrconv_fused_kernel(
    const float* __restrict__ xT,     // (B, N, C)
    const int*   __restrict__ edge,   // (2, B, N, K)
    const float* __restrict__ Wx,     // (OUT, C)
    const float* __restrict__ Wj,     // (OUT, C)
    const float* __restrict__ bias,   // (OUT)
    float*       __restrict__ out) {  // (B, OUT, N)
  __shared__ float hx[TN * HS2];      // node-major x tile:      hx[node][c]
  __shared__ float hj[TN * HS2];      // node-major max-rel tile hj[node][c]

  const int b    = blockIdx.y;
  const int n0   = blockIdx.x * TN;
  const int tid  = threadIdx.x;
  const int node = tid >> 3;          // 0..31
  const int ln8  = tid & 7;           // 0..7
  const float* xb = xT + (size_t)b * N_ * C_;

  // ---- Phase A: kick off async copy of the contiguous x tile into LDS. ----
  // Row 'node' of the tile: xT[b][n0+node][0:64] -> hx[node][0:64].
  // Each of the 8 threads per node moves 32B = 2 x b128, tracked on ASYNCcnt.
  {
    const float* gsrc = xb + (size_t)(n0 + node) * C_ + ln8 * 8;
    float*       ldst = &hx[node * HS2 + ln8 * 8];
    g_v4i_p gp = (g_v4i_p)(uintptr_t)gsrc;           // integer round-trip: drops const,
    l_v4i_p lp = (l_v4i_p)(uint32_t)(uintptr_t)ldst; // low 32 bits = LDS offset
    __builtin_amdgcn_global_load_async_to_lds_b128(gp, lp, 0, 0);
    __builtin_amdgcn_global_load_async_to_lds_b128(gp, lp, 16, 0);
  }

  // ---- Phase B (overlapped with async copy): gather-max over K neighbors. ----
  {
    const int c0 = ln8 * 8;
    const int n  = n0 + node;
    const int* E0 = edge + ((0 * B_ + b) * N_ + n) * K_;  // x_j indices
    const int* E1 = edge + ((1 * B_ + b) * N_ + n) * K_;  // x_i indices

    float mx[8];
#pragma unroll
    for (int cc = 0; cc < 8; ++cc) mx[cc] = -3.402823466e38f;

#pragma unroll 4
    for (int k = 0; k < K_; ++k) {
      const int j = E0[k];
      const int i = E1[k];
      const float4* pj = reinterpret_cast<const float4*>(xb + j * C_ + c0);
      const float4* pi = reinterpret_cast<const float4*>(xb + i * C_ + c0);
      float4 j0 = pj[0], j1 = pj[1];
      float4 i0 = pi[0], i1 = pi[1];
      mx[0] = fmaxf(mx[0], j0.x - i0.x);
      mx[1] = fmaxf(mx[1], j0.y - i0.y);
      mx[2] = fmaxf(mx[2], j0.z - i0.z);
      mx[3] = fmaxf(mx[3], j0.w - i0.w);
      mx[4] = fmaxf(mx[4], j1.x - i1.x);
      mx[5] = fmaxf(mx[5], j1.y - i1.y);
      mx[6] = fmaxf(mx[6], j1.z - i1.z);
      mx[7] = fmaxf(mx[7], j1.w - i1.w);
    }

    float4* q = reinterpret_cast<float4*>(&hj[node * HS2 + c0]);
    q[0] = make_float4(mx[0], mx[1], mx[2], mx[3]);
    q[1] = make_float4(mx[4], mx[5], mx[6], mx[7]);
  }

  // Drain this wave's async copies, then make LDS visible block-wide.
  asm volatile("s_wait_asynccnt 0x0" ::: "memory");
  __syncthreads();

  // ---- Phase C: D(16x16) = A(16x4) x B(4x16) + C; 8 waves x 2 column tiles. ----
  const int wave    = tid >> 5;
  const int lid     = tid & 31;
  const int m       = lid & 15;       // A row / B,D column
  const int hi      = lid >> 4;       // lane half selects K (A/B) and M (D) sub-range
  const int rowbase = wave * 16;      // this wave's 16 output rows

  v8f acc0, acc1;                     // bias folded into initial C accumulator
#pragma unroll
  for (int v = 0; v < 8; ++v) {       // D layout: M = v + 8*hi, N = m
    const float bv = bias[rowbase + 8 * hi + v];
    acc0[v] = bv;
    acc1[v] = bv;
  }

  const float* WxR = Wx + (rowbase + m) * C_;
  const float* WjR = Wj + (rowbase + m) * C_;

#pragma unroll 4
  for (int kk = 0; kk < 16; ++kk) {   // Wx @ hx, K = 64
    const int kb = 4 * kk + 2 * hi;   // fragment: K = vgpr + 2*hi
    v2f a;
    a.x = WxR[kb + 0];
    a.y = WxR[kb + 1];
    v2f b0 = *reinterpret_cast<const v2f*>(&hx[(m)      * HS2 + kb]);
    v2f b1 = *reinterpret_cast<const v2f*>(&hx[(16 + m) * HS2 + kb]);
    acc0 = __builtin_amdgcn_wmma_f32_16x16x4_f32(false, a, false, b0, (short)0,
                                                 acc0, false, false);
    acc1 = __builtin_amdgcn_wmma_f32_16x16x4_f32(false, a, false, b1, (short)0,
                                                 acc1, false, false);
  }

#pragma unroll 4
  for (int kk = 0; kk < 16; ++kk) {   // Wj @ hj, K = 64
    const int kb = 4 * kk + 2 * hi;
    v2f a;
    a.x = WjR[kb + 0];
    a.y = WjR[kb + 1];
    v2f b0 = *reinterpret_cast<const v2f*>(&hj[(m)      * HS2 + kb]);
    v2f b1 = *reinterpret_cast<const v2f*>(&hj[(16 + m) * HS2 + kb]);
    acc0 = __builtin_amdgcn_wmma_f32_16x16x4_f32(false, a, false, b0, (short)0,
                                                 acc0, false, false);
    acc1 = __builtin_amdgcn_wmma_f32_16x16x4_f32(false, a, false, b1, (short)0,
                                                 acc1, false, false);
  }

  // ---- relu + store: out[b][row][n] ----
  float* ob = out + (size_t)(b * OUT_) * N_ + n0;
#pragma unroll
  for (int v = 0; v < 8; ++v) {
    const int row = rowbase + 8 * hi + v;
    ob[row * N_ + m]      = fmaxf(acc0[v], 0.0f);
    ob[row * N_ + 16 + m] = fmaxf(acc1[v], 0.0f);
  }
}

extern "C" void kernel_launch(void* const* d_in, const int* in_sizes, int n_in,
                              void* d_out, int out_size, void* d_ws, size_t ws_size,
                              hipStream_t stream) {
  const float* x    = (const float*)d_in[0];   // (B, C, N, 1) f32
  const int*   edge = (const int*)  d_in[1];   // (2, B, N, K) i32
  const float* W    = (const float*)d_in[2];   // (OUT, 2C)   f32
  const float* bias = (const float*)d_in[3];   // (OUT)       f32
  float*       out  = (float*)d_out;           // (B, OUT, N, 1) f32

  float* xT = (float*)d_ws;                    // (B, N, C): 8 MB
  float* Wx = xT + (size_t)B_ * N_ * C_;       // (OUT, C):  32 KB
  float* Wj = Wx + (size_t)OUT_ * C_;          // (OUT, C):  32 KB

  wprep_kernel<<<(OUT_ * C_) / 256, 256, 0, stream>>>(W, Wx, Wj);

  dim3 tb(32, 8);
  dim3 tg(N_ / 32, C_ / 32, B_);
  xpose_kernel<<<tg, tb, 0, stream>>>(x, xT);

  dim3 g2(N_ / TN, B_);
  mrconv_fused_kernel<<<g2, 256, 0, stream>>>(xT, edge, Wx, Wj, bias, out);
}